// Propagator_16716012716114
// MI455X (gfx1250) — compile-verified
//
#include <hip/hip_runtime.h>
#include <math.h>

// ---------------------------------------------------------------------------
// Problem constants
// ---------------------------------------------------------------------------
#define NAO    384
#define NELEC  96
#define NFIELD 512
#define NTS    10
#define NP     (NAO * NELEC)      // 36864 elements per plane
#define NMAT   (NAO * NAO)        // 147456 elements per operator matrix
#define ORDER  6

typedef float v2f __attribute__((ext_vector_type(2)));
typedef float v8f __attribute__((ext_vector_type(8)));

// ---------------------------------------------------------------------------
// Small setup kernels
// ---------------------------------------------------------------------------

// tsh[i] = 0.5*(ts_v[i-1]+ts_v[i]) with zero-padded edges (i = 0..NTS)
// lw[0] = enuc (real part), lw[1] = 0 (imag part)
__global__ void precompute_kernel(const float* __restrict__ ts_v,
                                  const float* __restrict__ enuc,
                                  float* __restrict__ tsh,
                                  float* __restrict__ lw) {
  int i = threadIdx.x;
  if (i <= NTS) {
    float a = (i > 0)   ? ts_v[i - 1] : 0.0f;
    float b = (i < NTS) ? ts_v[i]     : 0.0f;
    tsh[i] = 0.5f * (a + b);
  }
  if (i == 0) { lw[0] = enuc[0]; lw[1] = 0.0f; }
}

// w_r = wfn, w_i = 0
__global__ void init_w_kernel(const float* __restrict__ wfn,
                              float* __restrict__ wR, float* __restrict__ wI) {
  int i = blockIdx.x * blockDim.x + threadIdx.x;
  if (i < NP) { wR[i] = wfn[i]; wI[i] = 0.0f; }
}

// term = cur ; acc = cur   (both planes)
__global__ void taylor_init_kernel(const float* __restrict__ cR, const float* __restrict__ cI,
                                   float* __restrict__ tR, float* __restrict__ tI,
                                   float* __restrict__ aR, float* __restrict__ aI) {
  int i = blockIdx.x * blockDim.x + threadIdx.x;
  if (i < NP) {
    float r = cR[i], im = cI[i];
    tR[i] = r;  tI[i] = im;
    aR[i] = r;  aI[i] = im;
  }
}

// ---------------------------------------------------------------------------
// build_v: M_t = sqrt(ts_v[t]) * sum_k fields[t,k] * vhs[k]   (t = 0..9)
// One pass over vhs (302 MB -> ~13us at 23.3 TB/s).  Implemented as a
// (16x512)@(512x147456) GEMM with V_WMMA_F32_16X16X4_F32, M padded 10->16.
// One wave produces a 16(t) x 16(col) output tile over K=512.
// ---------------------------------------------------------------------------
__global__ void build_v_kernel(const float* __restrict__ fields,  // (NTS, NFIELD)
                               const float* __restrict__ vhs,     // (NFIELD, NMAT)
                               const float* __restrict__ ts_v,
                               float* __restrict__ Vsum) {        // (NTS, NMAT)
  const int lane = threadIdx.x & 31;
  const int wave = threadIdx.x >> 5;
  const int g    = blockIdx.x * (blockDim.x >> 5) + wave;   // 0 .. 9215
  const long col0 = (long)g * 16;
  const int half = lane >> 4;
  const int l16  = lane & 15;
  const int m    = l16;                                     // padded timestep row
  const float mask = (m < NTS) ? 1.0f : 0.0f;
  const int mr = (m < NTS) ? m : (NTS - 1);

  v8f c = {0.f, 0.f, 0.f, 0.f, 0.f, 0.f, 0.f, 0.f};
  for (int kk = 0; kk < NFIELD; kk += 4) {
    const int k0 = kk + 2 * half;
    v2f a;
    a.x = fields[mr * NFIELD + k0]     * mask;
    a.y = fields[mr * NFIELD + k0 + 1] * mask;
    v2f b;
    b.x = vhs[(long)k0       * NMAT + col0 + l16];
    b.y = vhs[(long)(k0 + 1) * NMAT + col0 + l16];
    // stream-ahead hint (lowers to global_prefetch_b8)
    if (kk + 18 < NFIELD)
      __builtin_prefetch(vhs + (long)(k0 + 16) * NMAT + col0 + l16, 0, 1);
    c = __builtin_amdgcn_wmma_f32_16x16x4_f32(false, a, false, b,
                                              (short)0, c, false, false);
  }
#pragma unroll
  for (int r = 0; r < 8; ++r) {
    int t = r + 8 * half;                // C layout: VGPR r -> M=r | M=8+r
    if (t < NTS) {
      float s = sqrtf(ts_v[t]);          // |i*sqrt(dt)| ; the i is handled in mode 1
      Vsum[(long)t * NMAT + col0 + l16] = c[r] * s;
    }
  }
}

// ---------------------------------------------------------------------------
// One Taylor step of expm_apply:  term_k = (A_full @ term_{k-1}) / k,
// acc += term_k, for the two real planes of a complex wavefunction.
//   mode 0: A_full = alpha*A (real)      -> out_p = alpha*(A@in_p)/k
//   mode 1: A_full = i*A     (alpha=1)   -> out_r = -(A@in_i)/k, out_i = (A@in_r)/k
// alpha = -tsh[*] when alphaPtr != null (neg_ts_h), else 1.
// Grid: 36 blocks x 256 threads = 288 waves = 2 planes x 24x6 16x16 tiles.
// ---------------------------------------------------------------------------
__global__ void gemm_expm_step_kernel(const float* __restrict__ A,       // (NAO,NAO)
                                      const float* __restrict__ alphaPtr,
                                      float invk, int mode,
                                      const float* __restrict__ inR,
                                      const float* __restrict__ inI,
                                      float* __restrict__ outR,
                                      float* __restrict__ outI,
                                      float* __restrict__ accR,
                                      float* __restrict__ accI) {
  const int lane  = threadIdx.x & 31;
  const int wave  = threadIdx.x >> 5;
  const int g     = blockIdx.x * (blockDim.x >> 5) + wave;  // 0..287
  const int plane = g / 144;
  const int tile  = g % 144;
  const int tm = tile / 6, tn = tile % 6;

  const float alpha = alphaPtr ? -alphaPtr[0] : 1.0f;
  const float* B;
  float sgn;
  if (plane == 0) { B = mode ? inI : inR; sgn = mode ? -1.0f : 1.0f; }
  else            { B = mode ? inR : inI; sgn = 1.0f; }
  float* Dout = plane ? outI : outR;
  float* Dacc = plane ? accI : accR;

  const int half = lane >> 4;
  const int l16  = lane & 15;
  const int arow = tm * 16 + l16;
  const int bcol = tn * 16 + l16;
  const float* Arow = A + (long)arow * NAO;

  v8f c = {0.f, 0.f, 0.f, 0.f, 0.f, 0.f, 0.f, 0.f};
  for (int kk = 0; kk < NAO; kk += 4) {
    const int k0 = kk + 2 * half;       // A 16x4 layout: VGPR0={K0|K2}, VGPR1={K1|K3}
    v2f a;
    a.x = Arow[k0];
    a.y = Arow[k0 + 1];
    v2f b;
    b.x = B[(long)k0       * NELEC + bcol];
    b.y = B[(long)(k0 + 1) * NELEC + bcol];
    c = __builtin_amdgcn_wmma_f32_16x16x4_f32(false, a, false, b,
                                              (short)0, c, false, false);
  }

  const float s = alpha * invk * sgn;
#pragma unroll
  for (int r = 0; r < 8; ++r) {
    int row = tm * 16 + r + 8 * half;
    long idx = (long)row * NELEC + bcol;
    float v = c[r] * s;
    Dout[idx] = v;
    Dacc[idx] += v;
  }
}

// ---------------------------------------------------------------------------
// Per-column 2-norm over the orbital axis; scale column; record log(norm).
// Deterministic LDS tree reduction (no float atomics).
// ---------------------------------------------------------------------------
__global__ void colnorm_kernel(float* __restrict__ wR, float* __restrict__ wI,
                               float* __restrict__ lognorm) {
  __shared__ float red[128];
  const int c = blockIdx.x;                       // 0..95
  float s = 0.0f;
  for (int r = threadIdx.x; r < NAO; r += 128) {
    float a = wR[(long)r * NELEC + c];
    float b = wI[(long)r * NELEC + c];
    s += a * a + b * b;
  }
  red[threadIdx.x] = s;
  __syncthreads();
  for (int off = 64; off > 0; off >>= 1) {
    if (threadIdx.x < off) red[threadIdx.x] += red[threadIdx.x + off];
    __syncthreads();
  }
  const float norm = sqrtf(red[0]);
  const float inv  = 1.0f / norm;
  for (int r = threadIdx.x; r < NAO; r += 128) {
    wR[(long)r * NELEC + c] *= inv;
    wI[(long)r * NELEC + c] *= inv;
  }
  if (threadIdx.x == 0) lognorm[c] = logf(norm);
}

// lw_re += sum(lognorm) — single-thread, deterministic order
__global__ void lw_accum_kernel(const float* __restrict__ lognorm,
                                float* __restrict__ lw) {
  if (threadIdx.x == 0 && blockIdx.x == 0) {
    float s = 0.0f;
    for (int c = 0; c < NELEC; ++c) s += lognorm[c];
    lw[0] += s;
  }
}

// out = w * exp(i*imag(lw)/nelec) interleaved (re,im); out[2*NP] = real(lw)
__global__ void finalize_kernel(const float* __restrict__ wR,
                                const float* __restrict__ wI,
                                const float* __restrict__ lw,
                                float* __restrict__ out) {
  const float theta = lw[1] * (1.0f / (float)NELEC);
  const float cs = cosf(theta), sn = sinf(theta);
  for (int i = blockIdx.x * blockDim.x + threadIdx.x; i < NP;
       i += gridDim.x * blockDim.x) {
    float a = wR[i], b = wI[i];
    out[2 * i]     = a * cs - b * sn;
    out[2 * i + 1] = a * sn + b * cs;
  }
  if (blockIdx.x == 0 && threadIdx.x == 0) out[2 * NP] = lw[0];
}

// ---------------------------------------------------------------------------
// Host-side orchestration (graph-capture safe: only stream launches)
// ---------------------------------------------------------------------------
extern "C" void kernel_launch(void* const* d_in, const int* in_sizes, int n_in,
                              void* d_out, int out_size, void* d_ws, size_t ws_size,
                              hipStream_t stream) {
  const float* wfn    = (const float*)d_in[0];   // (384, 96)
  const float* fields = (const float*)d_in[1];   // (10, 512)
  const float* hmf    = (const float*)d_in[2];   // (384, 384)
  const float* vhs    = (const float*)d_in[3];   // (512, 384, 384)
  const float* ts_v   = (const float*)d_in[4];   // (10,)
  const float* enuc   = (const float*)d_in[5];   // (1,)
  float* out = (float*)d_out;
  float* ws  = (float*)d_ws;

  // Workspace carve-up (floats): ~7.1 MB total
  float* Vsum    = ws;                           // NTS * NMAT
  float* tsh     = Vsum + (long)NTS * NMAT;      // 16
  float* lw      = tsh + 16;                     // 2 (padded)
  float* lognorm = lw + 16;                      // 96 (padded)
  float* planes  = lognorm + 128;                // 8 * NP  (4 complex buffers)
  float* P[4][2];
  for (int i = 0; i < 4; ++i) {
    P[i][0] = planes + (long)(2 * i) * NP;
    P[i][1] = planes + (long)(2 * i + 1) * NP;
  }

  precompute_kernel<<<1, 32, 0, stream>>>(ts_v, enuc, tsh, lw);
  init_w_kernel<<<(NP + 255) / 256, 256, 0, stream>>>(wfn, P[0][0], P[0][1]);
  // Dominant-cost kernel: one streaming pass over vhs (302 MB) with f32 WMMA
  build_v_kernel<<<(NMAT / 16) / 8, 256, 0, stream>>>(fields, vhs, ts_v, Vsum);

  int cur = 0;
  auto expm = [&](const float* A, const float* alphaPtr, int mode) {
    int bufs[3], nb = 0;
    for (int i = 0; i < 4; ++i) if (i != cur) bufs[nb++] = i;
    const int tA = bufs[0], tB = bufs[1], acc = bufs[2];
    taylor_init_kernel<<<(NP + 255) / 256, 256, 0, stream>>>(
        P[cur][0], P[cur][1], P[tA][0], P[tA][1], P[acc][0], P[acc][1]);
    int in = tA, ob = tB;
    for (int k = 1; k <= ORDER; ++k) {
      gemm_expm_step_kernel<<<36, 256, 0, stream>>>(
          A, alphaPtr, 1.0f / (float)k, mode,
          P[in][0], P[in][1], P[ob][0], P[ob][1], P[acc][0], P[acc][1]);
      int t = in; in = ob; ob = t;
    }
    cur = acc;
  };

  for (int t = 0; t < NTS; ++t) {
    expm(hmf, tsh + t, /*mode=*/0);                       // expm(-ts_h[t]*hmf) @ w
    expm(Vsum + (long)t * NMAT, nullptr, /*mode=*/1);     // expm(i*M_t) @ w
    colnorm_kernel<<<NELEC, 128, 0, stream>>>(P[cur][0], P[cur][1], lognorm);
    lw_accum_kernel<<<1, 32, 0, stream>>>(lognorm, lw);
  }
  expm(hmf, tsh + NTS, /*mode=*/0);                       // final half step
  finalize_kernel<<<144, 256, 0, stream>>>(P[cur][0], P[cur][1], lw, out);
}